// CGCNNLayer_40965398069685
// MI455X (gfx1250) — compile-verified
//
#include <hip/hip_runtime.h>
#include <math.h>

typedef __attribute__((ext_vector_type(2))) float v2f;
typedef __attribute__((ext_vector_type(8))) float v8f;

constexpr int N_NODES = 50000;
constexpr int N_EDGES = 800000;
constexpr int AF   = 64;
constexpr int DIN  = 192;   // 2*AF + NBR_F
constexpr int DOUT = 128;
constexpr int TILES = N_EDGES / 16;   // 50000 (exact)
constexpr int KT = DIN / 4;           // 48 k-steps of the 16x16x4 WMMA
constexpr int NT = DOUT / 16;         // 8 n-tiles (== waves per block)
constexpr int FSTRIDE = 196;          // padded feature row stride (bank-conflict free b64)
constexpr int PFE = (16 * DIN) / 256; // 12 prefetch elements per thread
constexpr float EPSV = 1e-5f;

// workspace layout (float offsets)
constexpr int WS_BN1_SUM = 0;
constexpr int WS_BN1_SQ  = 128;
constexpr int WS_A1      = 256;
constexpr int WS_B1      = 384;
constexpr int WS_BN2_SUM = 512;
constexpr int WS_BN2_SQ  = 576;
constexpr int WS_A2      = 640;
constexpr int WS_B2      = 704;
constexpr int WS_UPD     = 768;                  // atom_update [N_NODES][AF]
constexpr int WS_TOTAL   = WS_UPD + N_NODES * AF;

__device__ __forceinline__ float softplus_f(float x) {
    return fmaxf(x, 0.f) + log1pf(__expf(-fabsf(x)));
}
__device__ __forceinline__ float sigmoid_f(float x) {
    return 1.f / (1.f + __expf(-x));
}

// gather one element of the concatenated per-edge feature row
__device__ __forceinline__ float gather_feat(const float* __restrict__ atom,
                                             const float* __restrict__ nbr,
                                             const int*   __restrict__ esrc,
                                             const int*   __restrict__ edst,
                                             int ebase, int e, int j) {
    if (j < AF)          return atom[esrc[ebase + e] * AF + j];
    else if (j < 2 * AF) return atom[edst[ebase + e] * AF + (j - AF)];
    else                 return nbr[(ebase + e) * AF + (j - 2 * AF)];
}

__global__ void k_zero(float* __restrict__ ws) {
    int i = blockIdx.x * blockDim.x + threadIdx.x;
    int stride = gridDim.x * blockDim.x;
    for (; i < WS_TOTAL; i += stride) ws[i] = 0.f;
}

// MODE 0: z = concat(a_src,a_dst,nbr) @ W + b, accumulate per-column sum / sumsq.
// MODE 1: recompute z, apply BN1 affine, msg = sigmoid(filter)*softplus(core),
//         scatter-add msg into atom_update via f32 global atomics.
template<int MODE>
__global__ __launch_bounds__(256)
void k_edge_gemm(const float* __restrict__ atom,
                 const float* __restrict__ nbr,
                 const int*   __restrict__ esrc,
                 const int*   __restrict__ edst,
                 const float* __restrict__ W,
                 const float* __restrict__ bias,
                 float* __restrict__ ws)
{
    __shared__ float sW[KT * NT * 64];      // 96 KB: fragment-packed weights
    __shared__ float sF[2][16 * FSTRIDE];   // double-buffered 16-edge feature tiles
    __shared__ float sZ[16 * 132];          // normalized activations (MODE 1)

    const int tid  = threadIdx.x;
    const int wave = tid >> 5;              // n-tile owned by this wave
    const int lane = tid & 31;
    const int half = lane >> 4;
    const int ln16 = lane & 15;
    const int col  = wave * 16 + ln16;      // output column this lane owns in C frags

    // Stage fragment-ordered W into LDS once per block.
    // Fragment f = kt*NT + nt; lane l holds (W[k][n], W[k+1][n]) with
    // k = kt*4 + 2*(l>>4), n = nt*16 + (l&15)  -- matches 16x16x4 B layout.
    for (int i = tid; i < KT * NT * 64; i += 256) {
        int f  = i >> 6;
        int r  = i & 63;
        int l  = r >> 1;
        int j  = r & 1;
        int kt = f >> 3;
        int nt = f & 7;
        int k  = kt * 4 + ((l >> 4) << 1) + j;
        int n  = nt * 16 + (l & 15);
        sW[i] = W[k * DOUT + n];
    }

    const float bcol = bias[col];
    float a1 = 0.f, b1 = 0.f;
    if (MODE == 1) {
        a1 = ws[WS_A1 + col];
        b1 = ws[WS_B1 + col];
    }
    float accS = 0.f, accQ = 0.f;
    float* __restrict__ upd = ws + WS_UPD;

    const float* wptr = &sW[wave * 64 + lane * 2];  // B frag base for this lane

    // Prologue: stage this block's first tile into buffer 0.
    {
        const int eb = blockIdx.x * 16;
        for (int i = tid; i < 16 * DIN; i += 256) {
            int e = i / DIN;
            int j = i - e * DIN;
            sF[0][e * FSTRIDE + j] = gather_feat(atom, nbr, esrc, edst, eb, e, j);
        }
    }
    __syncthreads();    // covers sW and sF[0]

    int cur = 0;
    for (int tile = blockIdx.x; tile < TILES; tile += gridDim.x) {
        const int ntile = tile + gridDim.x;

        // Warm the L2/L1 for the index rows two tiles ahead (global_prefetch_b8).
        const int ptile = tile + 2 * gridDim.x;
        if (ptile < TILES && tid < 16) {
            __builtin_prefetch(&esrc[ptile * 16 + tid], 0, 1);
            __builtin_prefetch(&edst[ptile * 16 + tid], 0, 1);
        }

        // Issue the gather for the NEXT tile now; latency overlaps the K-loop.
        float pre[PFE];
        if (ntile < TILES) {
            const int eb2 = ntile * 16;
            #pragma unroll
            for (int s = 0; s < PFE; ++s) {
                int i = tid + s * 256;
                int e = i / DIN;
                int j = i - e * DIN;
                pre[s] = gather_feat(atom, nbr, esrc, edst, eb2, e, j);
            }
        }

        v8f c;
        #pragma unroll
        for (int r = 0; r < 8; ++r) c[r] = bcol;   // bias folded into accumulator

        const float* fptr = &sF[cur][ln16 * FSTRIDE + half * 2];
        #pragma unroll
        for (int kt = 0; kt < KT; ++kt) {
            v2f a = *(const v2f*)(fptr + kt * 4);          // A[row][k..k+1]
            v2f b = *(const v2f*)(wptr + kt * (NT * 64));  // B[k..k+1][col]
            c = __builtin_amdgcn_wmma_f32_16x16x4_f32(
                    false, a, false, b, (short)0, c, false, false);
        }

        // Commit prefetched features to the other buffer (read by next iter
        // only after the end-of-iteration barrier).
        if (ntile < TILES) {
            float* nb = sF[cur ^ 1];
            #pragma unroll
            for (int s = 0; s < PFE; ++s) {
                int i = tid + s * 256;
                int e = i / DIN;
                int j = i - e * DIN;
                nb[e * FSTRIDE + j] = pre[s];
            }
        }

        if (MODE == 0) {
            #pragma unroll
            for (int r = 0; r < 8; ++r) { accS += c[r]; accQ += c[r] * c[r]; }
        } else {
            // C layout: reg r -> row (r + 8*half), col fixed per lane.
            const int rbase = half * 8;
            #pragma unroll
            for (int r = 0; r < 8; ++r)
                sZ[(rbase + r) * 132 + col] = a1 * c[r] + b1;
            __syncthreads();
            // gated message + scatter-add (16 rows x 64 cols, 4 per thread)
            const int ebase = tile * 16;
            for (int i = tid; i < 16 * AF; i += 256) {
                int row = i >> 6;
                int j   = i & 63;
                float fv = sZ[row * 132 + j];
                float gv = sZ[row * 132 + 64 + j];
                float m  = sigmoid_f(fv) * softplus_f(gv);
                atomicAdd(&upd[edst[ebase + row] * AF + j], m);
            }
        }

        __syncthreads();   // buffer swap / sZ reuse fence
        cur ^= 1;
    }

    if (MODE == 0) {
        atomicAdd(&ws[WS_BN1_SUM + col], accS);
        atomicAdd(&ws[WS_BN1_SQ  + col], accQ);
    }
}

__global__ void k_bn1_fin(const float* __restrict__ g, const float* __restrict__ b,
                          float* __restrict__ ws) {
    int j = threadIdx.x;                       // 128 threads
    float m = ws[WS_BN1_SUM + j] / (float)N_EDGES;
    float v = ws[WS_BN1_SQ + j] / (float)N_EDGES - m * m;
    float a = g[j] * rsqrtf(v + EPSV);
    ws[WS_A1 + j] = a;
    ws[WS_B1 + j] = b[j] - m * a;
}

__global__ void k_bn2_stats(float* __restrict__ ws) {
    const float* __restrict__ upd = ws + WS_UPD;
    int gth = blockIdx.x * blockDim.x + threadIdx.x;
    int col = gth & 63;
    int row0 = gth >> 6;
    int rstride = (gridDim.x * blockDim.x) >> 6;
    float s = 0.f, q = 0.f;
    for (int r = row0; r < N_NODES; r += rstride) {
        float v = upd[r * AF + col];
        s += v; q += v * v;
    }
    atomicAdd(&ws[WS_BN2_SUM + col], s);
    atomicAdd(&ws[WS_BN2_SQ  + col], q);
}

__global__ void k_bn2_fin(const float* __restrict__ g, const float* __restrict__ b,
                          float* __restrict__ ws) {
    int j = threadIdx.x;                       // 64 threads
    float m = ws[WS_BN2_SUM + j] / (float)N_NODES;
    float v = ws[WS_BN2_SQ + j] / (float)N_NODES - m * m;
    float a = g[j] * rsqrtf(v + EPSV);
    ws[WS_A2 + j] = a;
    ws[WS_B2 + j] = b[j] - m * a;
}

__global__ void k_final(const float* __restrict__ atom, const float* __restrict__ ws,
                        float* __restrict__ out) {
    const float* __restrict__ upd = ws + WS_UPD;
    int i = blockIdx.x * blockDim.x + threadIdx.x;
    int stride = gridDim.x * blockDim.x;
    for (; i < N_NODES * AF; i += stride) {
        int col = i & 63;
        float v = atom[i] + ws[WS_A2 + col] * upd[i] + ws[WS_B2 + col];
        out[i] = softplus_f(v);
    }
}

extern "C" void kernel_launch(void* const* d_in, const int* in_sizes, int n_in,
                              void* d_out, int out_size, void* d_ws, size_t ws_size,
                              hipStream_t stream)
{
    (void)in_sizes; (void)n_in; (void)out_size; (void)ws_size;
    const float* atom = (const float*)d_in[0];
    const float* nbr  = (const float*)d_in[1];
    const int*   esrc = (const int*)d_in[2];
    const int*   edst = (const int*)d_in[3];
    const float* W    = (const float*)d_in[4];
    const float* bias = (const float*)d_in[5];
    const float* g1   = (const float*)d_in[6];
    const float* b1   = (const float*)d_in[7];
    const float* g2   = (const float*)d_in[8];
    const float* b2   = (const float*)d_in[9];
    float* ws  = (float*)d_ws;
    float* out = (float*)d_out;

    k_zero<<<2048, 256, 0, stream>>>(ws);
    k_edge_gemm<0><<<512, 256, 0, stream>>>(atom, nbr, esrc, edst, W, bias, ws);
    k_bn1_fin<<<1, 128, 0, stream>>>(g1, b1, ws);
    k_edge_gemm<1><<<512, 256, 0, stream>>>(atom, nbr, esrc, edst, W, bias, ws);
    k_bn2_stats<<<256, 256, 0, stream>>>(ws);
    k_bn2_fin<<<1, 64, 0, stream>>>(g2, b2, ws);
    k_final<<<1024, 256, 0, stream>>>(atom, ws, out);
}